// GridLayer_88845693485419
// MI455X (gfx1250) — compile-verified
//
#include <hip/hip_runtime.h>

typedef __attribute__((ext_vector_type(4))) float float4v;

#define NH 9
#define FDIM 32
#define BVT 4
#define NODES_PER_BLOCK 32
#define THREADS 256
// staging: 32 nodes * 9 int32 = 288 ints = 1152 bytes = 72 lanes x b128

__global__ __launch_bounds__(THREADS) void gather_nh_kernel(
    const float* __restrict__ x,      // (BVT, N, F)
    const int*   __restrict__ adjc,   // (N, NH) int32
    float*       __restrict__ out,    // (BVT, N, NH, F)
    float*       __restrict__ outm,   // (N, NH) mask as 0/1 float
    int N)
{
    __shared__ int s_adj[NODES_PER_BLOCK * NH];   // 288 ints, LDS offset 0

    const int tid = threadIdx.x;
    const int n0  = blockIdx.x * NODES_PER_BLOCK;

    // ---- Stage this block's adjacency rows into LDS via the CDNA5 async
    // ---- global->LDS data mover (tracked by ASYNCcnt, no VGPR data).
    const int totalInts = NODES_PER_BLOCK * NH;        // 288
    const int lanes     = totalInts / 4;               // 72 lanes x 16B
    int validInts = (N - n0) * NH;
    if (validInts > totalInts) validInts = totalInts;
    if (tid < lanes && (tid * 4) < validInts) {
        const int* gsrc = adjc + (size_t)n0 * NH + tid * 4;
        // generic pointer to a __shared__ object: low 32 bits == LDS byte offset
        unsigned ldsOff = (unsigned)(size_t)(&s_adj[tid * 4]);
        asm volatile("global_load_async_to_lds_b128 %0, %1, off"
                     :: "v"(ldsOff), "v"(gsrc) : "memory");
    }
    asm volatile("s_wait_asynccnt 0" ::: "memory");
    __syncthreads();

    const int nn = tid >> 3;          // node within block [0,32)
    const int c  = tid & 7;           // float4 chunk [0,8)
    const int n  = n0 + nn;
    if (n >= N) return;

    const int j0raw = s_adj[nn * NH]; // self index (raw, per reference semantics)

    const size_t batchStrideX   = (size_t)N * FDIM;            // x:   per-b stride
    const size_t batchStrideOut = (size_t)N * NH * FDIM;       // out: per-b stride

    for (int nh = 0; nh < NH; ++nh) {
        int  jr      = s_adj[nn * NH + nh];
        bool invalid = (jr < 0) | (jr >= N);
        int  j       = invalid ? j0raw : jr;          // replace with adjc[:,0]
        j = (j < 0) ? 0 : ((j >= N) ? (N - 1) : j);   // jnp.take mode='clip'

        if (c == 0) {
            // mask is tiny (9.4 MB) -> also stream it non-temporally
            __builtin_nontemporal_store(invalid ? 1.0f : 0.0f,
                                        outm + (size_t)n * NH + nh);
        }

        const float* src0 = x   + (size_t)j * FDIM + c * 4;
        float*       dst0 = out + ((size_t)n * NH + nh) * FDIM + c * 4;

        #pragma unroll
        for (int b = 0; b < BVT; ++b) {
            // gather read: regular temporal hint -> x (128MB) stays resident in 192MB L2
            const float4v v = *(const float4v*)(src0 + (size_t)b * batchStrideX);
            // streaming write: non-temporal -> don't evict x from L2
            __builtin_nontemporal_store(v, (float4v*)(dst0 + (size_t)b * batchStrideOut));
        }
    }
}

extern "C" void kernel_launch(void* const* d_in, const int* in_sizes, int n_in,
                              void* d_out, int out_size, void* d_ws, size_t ws_size,
                              hipStream_t stream) {
    const float* x    = (const float*)d_in[0];
    const int*   adjc = (const int*)d_in[1];       // harness passes ints as int32
    const int    N    = in_sizes[1] / NH;          // 262144

    float* out  = (float*)d_out;
    float* outm = out + (size_t)BVT * N * NH * FDIM;  // mask follows x_nh, flat

    const int blocks = (N + NODES_PER_BLOCK - 1) / NODES_PER_BLOCK;  // 8192
    gather_nh_kernel<<<blocks, THREADS, 0, stream>>>(x, adjc, out, outm, N);
}